// Angles_52510270161244
// MI455X (gfx1250) — compile-verified
//
#include <hip/hip_runtime.h>
#include <math.h>

// Bond-angle kernel for MI455X (gfx1250, wave32).
//
// Memory-bound streaming gather:
//   - angles (int64 triples, 384 MB) -> non-temporal loads (don't evict L2)
//   - coords (12 MB) -> regular temporal gathers, stays resident in 192 MB L2
//   - output (64 MB) -> non-temporal stores (write-once, never re-read)
//   - grid-stride loop + global_prefetch_b8 of next iteration's index triple
//
// No WMMA: this workload has zero matrix structure; the CDNA5 features that
// matter here are the TH cache-policy field and the prefetch path.

#define BLOCK 256
#define GRID  16384   // 4M threads in flight; ~4 grid-stride iterations at N=16M

__global__ __launch_bounds__(BLOCK) void Angles_52510270161244_kernel(
    const float* __restrict__ coords,
    const long long* __restrict__ angles,
    float* __restrict__ out,
    int n)
{
    const long long stride = (long long)gridDim.x * (long long)blockDim.x;
    long long t = (long long)blockIdx.x * blockDim.x + threadIdx.x;

    for (; t < n; t += stride) {
        const long long* ap = angles + 3ll * t;

        // Software-pipeline the streaming index fetch for the next iteration.
        // Speculative prefetch: silently dropped if past end of allocation.
        __builtin_prefetch(ap + 3ll * stride, 0, 0);

        // Streaming triple: non-temporal so the 384 MB angle stream does not
        // evict the 12 MB coords working set from L2.
        long long i0 = __builtin_nontemporal_load(ap + 0);
        long long i1 = __builtin_nontemporal_load(ap + 1);
        long long i2 = __builtin_nontemporal_load(ap + 2);

        const float* ci = coords + 3ll * i0;
        const float* cj = coords + 3ll * i1;
        const float* ck = coords + 3ll * i2;

        // Gathers: regular temporal -> L2-resident (coords fits in L2).
        float jx = cj[0], jy = cj[1], jz = cj[2];
        float v1x = ci[0] - jx, v1y = ci[1] - jy, v1z = ci[2] - jz;
        float v2x = ck[0] - jx, v2y = ck[1] - jy, v2z = ck[2] - jz;

        float dot = v1x * v2x + v1y * v2y + v1z * v2z;
        float n1  = v1x * v1x + v1y * v1y + v1z * v1z;
        float n2  = v2x * v2x + v2y * v2y + v2z * v2z;

        // Match reference numerics: norm product via sqrt, true divide, clamp.
        float c = dot / sqrtf(n1 * n2);
        c = fminf(1.0f, fmaxf(-1.0f, c));
        float theta = acosf(c);

        __builtin_nontemporal_store(theta, out + t);
    }
}

extern "C" void kernel_launch(void* const* d_in, const int* in_sizes, int n_in,
                              void* d_out, int out_size, void* d_ws, size_t ws_size,
                              hipStream_t stream) {
    const float*     coords = (const float*)d_in[0];      // (N_ATOMS, 3) fp32
    const long long* angles = (const long long*)d_in[1];  // (N_ANGLES, 3) int64
    float*           out    = (float*)d_out;              // (N_ANGLES,) fp32

    int n = out_size;  // number of angles

    int blocks = (n + BLOCK - 1) / BLOCK;
    if (blocks > GRID) blocks = GRID;
    if (blocks < 1) blocks = 1;

    Angles_52510270161244_kernel<<<blocks, BLOCK, 0, stream>>>(coords, angles, out, n);
}